// SimpleMLPGen_1348619731539
// MI455X (gfx1250) — compile-verified
//
#include <hip/hip_runtime.h>
#include <cstdint>

// ---------------------------------------------------------------------------
// Hypernetwork MLP, fully fused so gen (2048 x 66112 = 541 MB) is never
// materialized. Both big einsums are GEMMs over an on-the-fly outer-product
// A matrix against gathered W_out tiles, using V_WMMA_F32_16X16X4_F32
// (native fp32 matrix path, wave32). W_out (135 MB) lives in the 192 MB L2;
// ~1000 FLOP/byte -> compute bound. B tiles are double-buffered through the
// CDNA5 async global->LDS path when the toolchain exposes it.
// ---------------------------------------------------------------------------

typedef __attribute__((ext_vector_type(2))) float v2f;
typedef __attribute__((ext_vector_type(8))) float v8f;
typedef __attribute__((ext_vector_type(2))) int v2i;

#define AS1 __attribute__((address_space(1)))
#define AS3 __attribute__((address_space(3)))

#if defined(__has_builtin)
#if __has_builtin(__builtin_amdgcn_global_load_async_to_lds_b64)
#define HAVE_ASYNC_LDS 1
#endif
#if __has_builtin(__builtin_amdgcn_s_wait_asynccnt)
#define HAVE_WAIT_ASYNC 1
#endif
#endif

__device__ __forceinline__ void wait_async0() {
#if defined(HAVE_ASYNC_LDS)
#if defined(HAVE_WAIT_ASYNC)
  __builtin_amdgcn_s_wait_asynccnt(0);
#else
  asm volatile("s_wait_asynccnt 0x0" ::: "memory");
#endif
#endif
}

__device__ __forceinline__ float elu1(float v) {
  return v > 0.0f ? v : __expf(v) - 1.0f;
}

__device__ __forceinline__ v8f wmma_f32_k4(v2f a, v2f b, v8f c) {
  // 8 args: (neg_a, A, neg_b, B, c_mod, C, reuse_a, reuse_b)
  return __builtin_amdgcn_wmma_f32_16x16x4_f32(false, a, false, b, (short)0, c,
                                               false, false);
}

// Stage (or async-issue) one 64n x 64k W_out tile into a [32 kpair][80] v2f
// buffer. 8 x b64 per thread, global reads coalesced along k.
__device__ __forceinline__ void stage_b_tile(const float* __restrict__ Wout,
                                             long rb, int k0, v2f* buf,
                                             int tid) {
#pragma unroll
  for (int q = 0; q < 8; ++q) {
    int p = tid + q * 256;
    int c2 = p & 31;  // kpair
    int n = p >> 5;   // column within tile
    const float* g = Wout + (size_t)(rb + n) * 512 + (k0 + c2 * 2);
    v2f* dst = buf + c2 * 80 + n;
#if defined(HAVE_ASYNC_LDS)
    uint32_t lo32 = (uint32_t)(uintptr_t)dst;
    __builtin_amdgcn_global_load_async_to_lds_b64(
        (AS1 v2i*)(uintptr_t)g, (AS3 v2i*)lo32, 0, 0);
#else
    *dst = *(const v2f*)g;
#endif
  }
}

// ---------------------------------------------------------------------------
// Stage 1: h[b,j] = elu( meta[b,:] . W_in[j,:] + b_in[j] )  (0.1% of FLOPs)
// ---------------------------------------------------------------------------
__global__ void hyper_hidden_kernel(const float* __restrict__ x,
                                    const float* __restrict__ W_in,
                                    const float* __restrict__ b_in,
                                    float* __restrict__ h) {
  int idx = blockIdx.x * blockDim.x + threadIdx.x; // [0, 2048*512)
  int b = idx >> 9;
  int j = idx & 511;
  const float* xm = x + b * 128 + 64; // meta = x[:, 64:128]
  const float* wr = W_in + j * 64;
  float acc = b_in[j];
#pragma unroll 8
  for (int m = 0; m < 64; ++m) acc = fmaf(xm[m], wr[m], acc);
  h[idx] = elu1(acc);
}

// ---------------------------------------------------------------------------
// Unified generated-weight GEMM:
//   out[b,n] = sum_{io} scale'[b,io] * ( sum_k h[b,k]*Wout[rowBase+io*rowStride+n,k]
//                                        + bOut[row] )
// scale'[b,scaleVirt]==1 folds the generated-bias ROWS of W_out; the bOut
// term is folded as one virtual K-step per io.
//
// 256 thr = 8 wave32. WG tile 256(b) x 64(n); per wave 2 M-subtiles x 4
// N-subtiles = 8 accumulators -> 6 ds_load_b64 per 8 WMMA. LDS tiles are v2f
// [kpair][...] so every fragment is one naturally-paired b64 (pads 272/80:
// half-wave offset = 32 banks -> conflict-free). B tile is double-buffered
// via async global->LDS; grid.z slices split the outer loop (deterministic
// split-K into workspace partials).
// ---------------------------------------------------------------------------
__global__ __launch_bounds__(256) void hyper_gen_gemm(
    const float* __restrict__ Wout, const float* __restrict__ bOut,
    const float* __restrict__ hbuf, const float* __restrict__ scale,
    float* __restrict__ outBase, int scaleStride, int scaleVirt, long rowBase,
    int rowStride, int outerTotal, int outerChunk, int ldOut,
    long sliceStride) {
  __shared__ v2f lds_h2[32][272];   // [kpair][row 0..255 + pad]
  __shared__ v2f lds_b2[2][32][80]; // double-buffered [kpair][n + pad]

  const int tid = threadIdx.x;
  const int lane = tid & 31;
  const int wave = tid >> 5;
  const int lo = lane & 15;
  const int hi = lane >> 4;

  const int n0 = blockIdx.x * 64;
  const int rowBlock = blockIdx.y * 256;
  const int i0 = blockIdx.z * outerChunk;
  int i1 = i0 + outerChunk;
  if (i1 > outerTotal) i1 = outerTotal;

  const int rowA = wave * 16 + lo;      // local row, M-subtile 0
  const int gRow0 = rowBlock + rowA;    // global batch rows
  const int gRow1 = gRow0 + 128;

  v8f zero = {};
  v8f acc0[4] = {zero, zero, zero, zero};
  v8f acc1[4] = {zero, zero, zero, zero};

  for (int k0 = 0; k0 < 512; k0 += 64) {
    __syncthreads(); // prior chunk's readers done -> safe to restage
    // h-tile: 256 rows x 32 kpairs, coalesced float2 rows.
    for (int p = tid; p < 256 * 32; p += 256) {
      int c2 = p & 31, r = p >> 5;
      lds_h2[c2][r] =
          *(const v2f*)(hbuf + (size_t)(rowBlock + r) * 512 + k0 + c2 * 2);
    }
    int cur = 0;
    stage_b_tile(Wout, rowBase + (long)i0 * rowStride + n0, k0,
                 &lds_b2[0][0][0], tid);
    for (int io = i0; io < i1; ++io) {
      wait_async0();   // this wave's async writes to buf[cur] landed
      __syncthreads(); // all waves wrote buf[cur]; prior readers of buf[nxt] done
      if (io + 1 < i1)
        stage_b_tile(Wout, rowBase + (long)(io + 1) * rowStride + n0, k0,
                     &lds_b2[cur ^ 1][0][0], tid);
      float rs0 =
          (io < scaleVirt) ? scale[(size_t)gRow0 * scaleStride + io] : 1.0f;
      float rs1 =
          (io < scaleVirt) ? scale[(size_t)gRow1 * scaleStride + io] : 1.0f;
#pragma unroll 4
      for (int ks = 0; ks < 16; ++ks) {
        int kp = ks * 2 + hi; // this lane's kpair within the 4-wide K step
        v2f a0 = rs0 * lds_h2[kp][rowA];
        v2f a1 = rs1 * lds_h2[kp][rowA + 128];
#pragma unroll
        for (int nt = 0; nt < 4; ++nt) {
          v2f bb = lds_b2[cur][kp][nt * 16 + lo];
          acc0[nt] = wmma_f32_k4(a0, bb, acc0[nt]);
          acc1[nt] = wmma_f32_k4(a1, bb, acc1[nt]);
        }
      }
      cur ^= 1;
    }
  }

  // Generated-bias fold: one virtual K-step per io (h-column == 1, B row ==
  // bOut[row]); only virtual K=0 (lanes<16) carries data.
  for (int io = i0; io < i1; ++io) {
    float rs0 =
        (io < scaleVirt) ? scale[(size_t)gRow0 * scaleStride + io] : 1.0f;
    float rs1 =
        (io < scaleVirt) ? scale[(size_t)gRow1 * scaleStride + io] : 1.0f;
    long rb = rowBase + (long)io * rowStride + n0;
    v2f a0, a1;
    a0[0] = hi ? 0.0f : rs0;
    a0[1] = 0.0f;
    a1[0] = hi ? 0.0f : rs1;
    a1[1] = 0.0f;
#pragma unroll
    for (int nt = 0; nt < 4; ++nt) {
      v2f bb;
      bb[0] = hi ? 0.0f : bOut[rb + nt * 16 + lo];
      bb[1] = 0.0f;
      acc0[nt] = wmma_f32_k4(a0, bb, acc0[nt]);
      acc1[nt] = wmma_f32_k4(a1, bb, acc1[nt]);
    }
  }

  // Epilogue: C/D layout -> element (M = jv + 8*hi, N = nt*16 + lo).
  float* outP = outBase + (size_t)blockIdx.z * sliceStride;
#pragma unroll
  for (int nt = 0; nt < 4; ++nt) {
#pragma unroll
    for (int jv = 0; jv < 8; ++jv) {
      int r0 = rowBlock + wave * 16 + jv + 8 * hi;
      int col = n0 + nt * 16 + lo;
      outP[(size_t)r0 * ldOut + col] = acc0[nt][jv];
      outP[(size_t)(r0 + 128) * ldOut + col] = acc1[nt][jv];
    }
  }
}

// ---------------------------------------------------------------------------
// Deterministic split-K reduction (no float atomics); optional fused ELU.
// ---------------------------------------------------------------------------
__global__ void hyper_reduce_kernel(const float* __restrict__ part,
                                    float* __restrict__ out, int nsl,
                                    long stride, int doElu) {
  int idx = blockIdx.x * blockDim.x + threadIdx.x;
  float s = 0.0f;
  for (int sl = 0; sl < nsl; ++sl) s += part[(size_t)sl * stride + idx];
  out[idx] = doElu ? elu1(s) : s;
}

extern "C" void kernel_launch(void* const* d_in, const int* in_sizes, int n_in,
                              void* d_out, int out_size, void* d_ws,
                              size_t ws_size, hipStream_t stream) {
  const float* x = (const float*)d_in[0];     // (2048, 128)
  const float* W_in = (const float*)d_in[1];  // (512, 64)
  const float* b_in = (const float*)d_in[2];  // (512,)
  const float* W_out = (const float*)d_in[3]; // (66112, 512)
  const float* b_out = (const float*)d_in[4]; // (66112,)
  float* out = (float*)d_out;                 // (2048, 64)

  float* ws = (float*)d_ws;
  float* hbuf = ws;                              // 2048*512           (4 MB)
  float* ybuf = hbuf + (size_t)2048 * 512;       // 2048*512           (4 MB)
  float* p1 = ybuf + (size_t)2048 * 512;         // 2 x 2048*512       (8 MB)
  float* p2 = p1 + (size_t)2 * 2048 * 512;       // 16 x 2048*64       (8 MB)

  hyper_hidden_kernel<<<(2048 * 512) / 256, 256, 0, stream>>>(x, W_in, b_in,
                                                              hbuf);

  // y_pre partials: outer i in [0,65): i<64 -> scale=real[b,i], rows i*512+h;
  // i==64 -> scale=1, rows 32768+h (= l_in_b block, contiguous by layout).
  hyper_gen_gemm<<<dim3(8, 8, 2), 256, 0, stream>>>(
      W_out, b_out, hbuf, x, p1,
      /*scaleStride*/ 128, /*scaleVirt*/ 64,
      /*rowBase*/ 0L, /*rowStride*/ 512,
      /*outerTotal*/ 65, /*outerChunk*/ 33,
      /*ldOut*/ 512, /*sliceStride*/ (long)2048 * 512);
  // y = elu(sum of 2 slices)
  hyper_reduce_kernel<<<(2048 * 512) / 256, 256, 0, stream>>>(
      p1, ybuf, 2, (long)2048 * 512, 1);

  // out partials: outer h in [0,513): h<512 -> scale=y[b,h], rows
  // 33280+h*64+o; h==512 -> scale=1, rows 66048+o (= l_out_b block).
  hyper_gen_gemm<<<dim3(1, 8, 16), 256, 0, stream>>>(
      W_out, b_out, hbuf, ybuf, p2,
      /*scaleStride*/ 512, /*scaleVirt*/ 512,
      /*rowBase*/ 33280L, /*rowStride*/ 64,
      /*outerTotal*/ 513, /*outerChunk*/ 33,
      /*ldOut*/ 64, /*sliceStride*/ (long)2048 * 64);
  hyper_reduce_kernel<<<(2048 * 64) / 256, 256, 0, stream>>>(
      p2, out, 16, (long)2048 * 64, 0);
}